// SupConLossGraph_11450382811333
// MI455X (gfx1250) — compile-verified
//
#include <hip/hip_runtime.h>
#include <hip/hip_bf16.h>

typedef __attribute__((ext_vector_type(2))) float v2f;
typedef __attribute__((ext_vector_type(8))) float v8f;

#define NUM_GEOM 15
#define NUM_POSS 11
#define NUM_SEM  24
#define R_TOTAL  50
#define R_PAD    64
#define EMBED    512
#define BATCH    65536
#define INV_T    (1.0f / 0.07f)
#define EPS_F    1e-7f
#define WAVES_PER_BLOCK 8
#define ROWS_PER_WAVE   16
#define ROWS_PER_BLOCK  (WAVES_PER_BLOCK * ROWS_PER_WAVE)

// Build zero-padded [64 x 512] table in workspace; zero the scalar output.
__global__ void supcon_prep_kernel(const float* __restrict__ g,
                                   const float* __restrict__ p,
                                   const float* __restrict__ s,
                                   float* __restrict__ tab,
                                   float* __restrict__ out) {
    int i = blockIdx.x * blockDim.x + threadIdx.x;   // 0 .. 64*512-1
    if (i == 0) out[0] = 0.0f;
    if (i >= R_PAD * EMBED) return;
    int row = i >> 9;          // /512
    int col = i & (EMBED - 1);
    float v = 0.0f;
    if (row < NUM_GEOM)                 v = g[row * EMBED + col];
    else if (row < NUM_GEOM + NUM_POSS) v = p[(row - NUM_GEOM) * EMBED + col];
    else if (row < R_TOTAL)             v = s[(row - NUM_GEOM - NUM_POSS) * EMBED + col];
    tab[i] = v;
}

__global__ __launch_bounds__(WAVES_PER_BLOCK * 32)
void supcon_main_kernel(const float* __restrict__ x,      // [B,512] raw embeds
                        const float* __restrict__ tab,    // [64,512] padded table
                        const int*   __restrict__ rel,    // [B] relation ids
                        float* __restrict__ out) {
    __shared__ float lds[WAVES_PER_BLOCK * ROWS_PER_WAVE * R_PAD];  // 32 KB

    const int lane = threadIdx.x & 31;
    const int wave = threadIdx.x >> 5;
    const int half = lane >> 4;      // 0: lanes 0-15, 1: lanes 16-31
    const int sub  = lane & 15;

    const long rowBase = ((long)blockIdx.x * WAVES_PER_BLOCK + wave) * ROWS_PER_WAVE;

    // A fragment (16x4 f32): lane L holds row (L&15), K-pair {4k+2*half, +1}
    const float* arow = x   + (rowBase + sub) * EMBED + 2 * half;
    // B fragment (4x16 f32): lane L holds col (L&15), K-pair {4k+2*half, +1}
    const float* brow = tab + (long)sub * EMBED + 2 * half;

    v8f acc0 = {}, acc1 = {}, acc2 = {}, acc3 = {};
    float ssq = 0.0f;

    #pragma unroll 4
    for (int k = 0; k < EMBED / 4; ++k) {
        __builtin_prefetch(arow + 4 * k + 128, 0, 1);  // global_prefetch 512B ahead
        v2f a  = *(const v2f*)(arow + 4 * k);
        ssq += a.x * a.x + a.y * a.y;
        v2f b0 = *(const v2f*)(brow + 4 * k);
        v2f b1 = *(const v2f*)(brow + 16 * EMBED + 4 * k);
        v2f b2 = *(const v2f*)(brow + 32 * EMBED + 4 * k);
        v2f b3 = *(const v2f*)(brow + 48 * EMBED + 4 * k);
        // D = A x B + C, full fp32 path (V_WMMA_F32_16X16X4_F32)
        acc0 = __builtin_amdgcn_wmma_f32_16x16x4_f32(false, a, false, b0, (short)0, acc0, false, false);
        acc1 = __builtin_amdgcn_wmma_f32_16x16x4_f32(false, a, false, b1, (short)0, acc1, false, false);
        acc2 = __builtin_amdgcn_wmma_f32_16x16x4_f32(false, a, false, b2, (short)0, acc2, false, false);
        acc3 = __builtin_amdgcn_wmma_f32_16x16x4_f32(false, a, false, b3, (short)0, acc3, false, false);
    }

    // Full row sum-of-squares: combine the two K-halves. Lane L then holds
    // ssq for row (L&15).
    ssq += __shfl_xor(ssq, 16, 32);
    const float scale = (1.0f / sqrtf(ssq)) * INV_T;   // folds L2-norm and 1/T

    // Spill raw dots to this wave's LDS region:
    // C/D layout: vgpr j <-> row j + 8*half ; lane sub <-> column t*16+sub
    float* my = lds + wave * (ROWS_PER_WAVE * R_PAD);
    #pragma unroll
    for (int j = 0; j < 8; ++j) {
        int r = (j + 8 * half) * R_PAD + sub;
        my[r +  0] = acc0[j];
        my[r + 16] = acc1[j];
        my[r + 32] = acc2[j];
        my[r + 48] = acc3[j];
    }
    __syncthreads();

    // Epilogue: lane = (row = sub, column half = half) handles 32 columns.
    const int   my_rel = rel[rowBase + sub];
    const int   scat   = (my_rel < NUM_GEOM) ? 0 : (my_rel < NUM_GEOM + NUM_POSS ? 1 : 2);
    const float nneg   = (scat == 0) ? (float)(NUM_GEOM - 1)
                       : (scat == 1) ? (float)(NUM_POSS - 1)
                                     : (float)(NUM_SEM - 1);

    const float* rowp = my + sub * R_PAD + half * 32;
    float esum = 0.0f, pos = 0.0f;
    #pragma unroll 8
    for (int c0 = 0; c0 < 32; ++c0) {
        int   c = half * 32 + c0;
        float d = rowp[c0] * scale;
        int  cc = (c < NUM_GEOM) ? 0
                : (c < NUM_GEOM + NUM_POSS) ? 1
                : (c < R_TOTAL) ? 2 : 3;          // 3 = padding, never matches
        if (c == my_rel)      pos  = d;
        else if (cc == scat)  esum += __expf(d);
    }
    esum += __shfl_xor(esum, 16, 32);
    pos  += __shfl_xor(pos,  16, 32);

    // (T/BASE_T)==1; contribution per sample
    float contrib = (__logf(__expf(pos) + EPS_F) - __logf(esum + EPS_F)) / (-nneg);
    if (lane >= 16) contrib = 0.0f;               // halves are duplicates

    #pragma unroll
    for (int off = 1; off < 32; off <<= 1)
        contrib += __shfl_xor(contrib, off, 32);
    if (lane == 0) atomicAdd(out, contrib);
}

extern "C" void kernel_launch(void* const* d_in, const int* in_sizes, int n_in,
                              void* d_out, int out_size, void* d_ws, size_t ws_size,
                              hipStream_t stream) {
    const float* x   = (const float*)d_in[0];   // predicted_txt_embeds [65536,512]
    const float* g   = (const float*)d_in[1];   // emb_geom [15,512]
    const float* p   = (const float*)d_in[2];   // emb_poss [11,512]
    const float* s   = (const float*)d_in[3];   // emb_sem  [24,512]
    const int*   rel = (const int*)d_in[4];     // curr_relation_ids [65536]
    float* out = (float*)d_out;
    float* tab = (float*)d_ws;                  // padded [64,512] table (128 KB)

    supcon_prep_kernel<<<(R_PAD * EMBED + 255) / 256, 256, 0, stream>>>(g, p, s, tab, out);

    const int grid = BATCH / ROWS_PER_BLOCK;    // 65536 / 128 = 512
    supcon_main_kernel<<<grid, WAVES_PER_BLOCK * 32, 0, stream>>>(x, tab, rel, out);
}